// SimGNN_85839216378399
// MI455X (gfx1250) — compile-verified
//
#include <hip/hip_runtime.h>
#include <hip/hip_bf16.h>

// ---------------- problem constants (match reference) ----------------
static constexpr int B = 64, N = 512, E = 8192;
static constexpr int F_IN = 128, F1 = 64, F2 = 32, F3 = 16;
static constexpr int K = 16, BINS = 16;
static constexpr int M1 = 32, M2 = 16, M3 = 8, M4 = 4;

typedef __attribute__((ext_vector_type(2))) float v2f;
typedef __attribute__((ext_vector_type(8))) float v8f;

// ---------------- helpers ----------------
__device__ __forceinline__ float sigmoidf_(float x) { return 1.0f / (1.0f + expf(-x)); }

// monotone float <-> orderable uint mapping for atomic min/max on floats
__device__ __forceinline__ unsigned fkey(float f) {
    unsigned u = __float_as_uint(f);
    return (u & 0x80000000u) ? ~u : (u | 0x80000000u);
}
__device__ __forceinline__ float funkey(unsigned k) {
    unsigned u = (k & 0x80000000u) ? (k & 0x7FFFFFFFu) : ~k;
    return __uint_as_float(u);
}

// ---------------- adjacency construction ----------------
__global__ void adj_scatter_kernel(const int* __restrict__ ei, float* __restrict__ A) {
    long t = (long)blockIdx.x * blockDim.x + threadIdx.x;
    long total = (long)B * (E + N);
    if (t >= total) return;
    int b = (int)(t / (E + N));
    int k = (int)(t % (E + N));
    if (k < E) {
        int r = ei[(long)b * 2 * E + k];
        int c = ei[(long)b * 2 * E + E + k];
        A[((long)b * N + r) * N + c] = 1.0f;
    } else {
        int n = k - E;
        A[((long)b * N + n) * N + n] = 1.0f;
    }
}

__global__ void rowsum_dinv_kernel(const float* __restrict__ A, float* __restrict__ dinv) {
    int t = blockIdx.x * blockDim.x + threadIdx.x;
    if (t >= B * N) return;
    const float* row = A + (long)t * N;
    float s = 0.0f;
    for (int c = 0; c < N; ++c) s += row[c];
    dinv[t] = rsqrtf(fmaxf(s, 1e-12f));
}

__global__ void adj_normalize_kernel(float* __restrict__ A, const float* __restrict__ dinv) {
    long t = (long)blockIdx.x * blockDim.x + threadIdx.x;
    long total = (long)B * N * N;
    if (t >= total) return;
    int b = (int)(t / ((long)N * N));
    long rem = t % ((long)N * N);
    int r = (int)(rem / N), c = (int)(rem % N);
    A[t] *= dinv[b * N + r] * dinv[b * N + c];
}

// ---------------- dense feature transform: out[b,n,:] = x[b,n,:] @ W ----------------
__global__ void xw_kernel(const float* __restrict__ x, const float* __restrict__ w,
                          float* __restrict__ out, int fin, int fout) {
    long t = (long)blockIdx.x * blockDim.x + threadIdx.x;
    long total = (long)B * N * fout;
    if (t >= total) return;
    long bn = t / fout;
    int j = (int)(t % fout);
    const float* xr = x + bn * fin;
    float acc = 0.0f;
    for (int i = 0; i < fin; ++i) acc += xr[i] * w[i * fout + j];
    out[t] = acc;
}

// ---------------- WMMA fp32 aggregation: Y = act(A @ H + bias) ----------------
// 8 waves / block, one row tile per wave; each wave register-blocks over ALL
// column tiles (NT = F/16) so every A fragment is loaded exactly once.
template <int NT>
__global__ void gcn_agg_wmma_kernel(const float* __restrict__ A, const float* __restrict__ H,
                                    const float* __restrict__ bias, float* __restrict__ Y,
                                    int do_relu) {
    constexpr int F = NT * 16;
    int lane = threadIdx.x & 31;
    int wave = threadIdx.x >> 5;                 // 0..7
    int half = lane >> 4, l = lane & 15;
    int row0 = (blockIdx.x * 8 + wave) * 16;
    int b = blockIdx.y;
    const float* Ab = A + (long)b * N * N;
    const float* Hb = H + (long)b * N * F;
    const float* Arow = Ab + (long)(row0 + l) * N;

    v8f zero = {};
    v8f acc[NT];
#pragma unroll
    for (int t = 0; t < NT; ++t) acc[t] = zero;

    for (int k0 = 0; k0 < N; k0 += 4) {
        if ((k0 & 28) == 0 && k0 + 32 < N)
            __builtin_prefetch(&Arow[k0 + 32], 0, 3);   // global_prefetch_b8, one line ahead
        int ka = k0 + 2 * half;
        v2f a;
        a.x = Arow[ka];
        a.y = Arow[ka + 1];
#pragma unroll
        for (int t = 0; t < NT; ++t) {
            int col0 = t * 16;
            v2f bv;
            bv.x = Hb[(long)ka * F + col0 + l];
            bv.y = Hb[(long)(ka + 1) * F + col0 + l];
            acc[t] = __builtin_amdgcn_wmma_f32_16x16x4_f32(false, a, false, bv,
                                                           (short)0, acc[t], false, false);
        }
    }
#pragma unroll
    for (int t = 0; t < NT; ++t) {
        int col0 = t * 16;
        float bcol = bias[col0 + l];
#pragma unroll
        for (int r = 0; r < 8; ++r) {
            int m = row0 + r + 8 * half;
            float v = acc[t][r] + bcol;
            if (do_relu) v = fmaxf(v, 0.0f);
            Y[((long)b * N + m) * F + col0 + l] = v;
        }
    }
}

// ---------------- attention pooling (mean commutes with the linear map) ----------------
__global__ void att_pool_kernel(const float* __restrict__ u, const float* __restrict__ attn_w,
                                float* __restrict__ h) {
    int b = blockIdx.x, tid = threadIdx.x;       // 256 threads
    __shared__ float sred[256];
    __shared__ float ubar[16];
    __shared__ float cvec[16];
    const float* ub = u + (long)b * N * F3;
    int d = tid & 15, grp = tid >> 4;
    float acc = 0.0f;
    for (int n = grp; n < N; n += 16) acc += ub[n * F3 + d];
    sred[tid] = acc;
    __syncthreads();
    for (int s = 8; s > 0; s >>= 1) {
        if (grp < s) sred[tid] += sred[tid + s * 16];
        __syncthreads();
    }
    if (tid < 16) ubar[tid] = sred[tid] * (1.0f / N);
    __syncthreads();
    if (tid < 16) {
        float a = 0.0f;
        for (int e = 0; e < 16; ++e) a += ubar[e] * attn_w[e * 16 + tid];
        cvec[tid] = tanhf(a);
    }
    __syncthreads();
    float hacc[16];
    for (int q = 0; q < 16; ++q) hacc[q] = 0.0f;
    for (int n = tid; n < N; n += 256) {
        const float* row = ub + n * F3;
        float dot = 0.0f;
        for (int e = 0; e < 16; ++e) dot += row[e] * cvec[e];
        float s = sigmoidf_(dot);
        for (int e = 0; e < 16; ++e) hacc[e] += s * row[e];
    }
    for (int q = 0; q < 16; ++q) {
        __syncthreads();
        sred[tid] = hacc[q];
        __syncthreads();
        for (int s = 128; s > 0; s >>= 1) {
            if (tid < s) sred[tid] += sred[tid + s];
            __syncthreads();
        }
        if (tid == 0) h[b * 16 + q] = sred[0];
    }
}

// ---------------- NTN interaction; lanes 16..31 zero the histogram slots ----------------
__global__ void ntn_kernel(const float* __restrict__ h_i, const float* __restrict__ h_j,
                           const float* __restrict__ ntn_w, const float* __restrict__ ntn_v,
                           const float* __restrict__ ntn_b, float* __restrict__ feat) {
    int b = blockIdx.x, lane = threadIdx.x;      // 32 threads
    __shared__ float hi[16], hj[16];
    if (lane < 16) { hi[lane] = h_i[b * 16 + lane]; hj[lane] = h_j[b * 16 + lane]; }
    __syncthreads();
    if (lane < K) {
        int k = lane;
        float bil = 0.0f;
        for (int d = 0; d < 16; ++d) {
            const float* wr = ntn_w + (k * 16 + d) * 16;
            float t = 0.0f;
            for (int e = 0; e < 16; ++e) t += wr[e] * hj[e];
            bil += hi[d] * t;
        }
        float aff = 0.0f;
        for (int d = 0; d < 16; ++d)
            aff += ntn_v[k * 32 + d] * hi[d] + ntn_v[k * 32 + 16 + d] * hj[d];
        feat[b * (K + BINS) + k] = tanhf(bil + aff + ntn_b[k]);
    } else {
        feat[b * (K + BINS) + lane] = 0.0f;
    }
}

// ---------------- similarity tile via WMMA (z = u_i @ u_j^T, 16x16 tile) ----------------
__device__ __forceinline__ v8f sim_tile(const float* __restrict__ uib,
                                        const float* __restrict__ ujb,
                                        int row0, int col0, int half, int l) {
    v8f c = {};
#pragma unroll
    for (int k0 = 0; k0 < F3; k0 += 4) {
        int ka = k0 + 2 * half;
        v2f a, bv;
        a.x = uib[(long)(row0 + l) * F3 + ka];
        a.y = uib[(long)(row0 + l) * F3 + ka + 1];
        bv.x = ujb[(long)(col0 + l) * F3 + ka];    // B = u_j^T
        bv.y = ujb[(long)(col0 + l) * F3 + ka + 1];
        c = __builtin_amdgcn_wmma_f32_16x16x4_f32(false, a, false, bv, (short)0, c, false, false);
    }
    return c;
}

__global__ void minmax_init_kernel(unsigned* __restrict__ mm) {
    int t = blockIdx.x * blockDim.x + threadIdx.x;
    if (t >= B) return;
    mm[t * 2 + 0] = 0xFFFFFFFFu;   // min slot
    mm[t * 2 + 1] = 0u;            // max slot
}

// pass 1: per-batch min/max of sigmoid(z); 8 waves per block, one col tile per wave
__global__ void sim_minmax_kernel(const float* __restrict__ ui, const float* __restrict__ uj,
                                  unsigned* __restrict__ mm) {
    int lane = threadIdx.x & 31, wave = threadIdx.x >> 5;
    int half = lane >> 4, l = lane & 15;
    int b = blockIdx.z;
    int row0 = blockIdx.x * 16;
    int col0 = (blockIdx.y * 8 + wave) * 16;
    const float* uib = ui + (long)b * N * F3;
    const float* ujb = uj + (long)b * N * F3;
    v8f c = sim_tile(uib, ujb, row0, col0, half, l);
    float lmin = 1e30f, lmax = -1e30f;
#pragma unroll
    for (int r = 0; r < 8; ++r) {
        float sg = sigmoidf_(c[r]);
        lmin = fminf(lmin, sg);
        lmax = fmaxf(lmax, sg);
    }
    for (int off = 16; off; off >>= 1) {
        lmin = fminf(lmin, __shfl_xor(lmin, off, 32));
        lmax = fmaxf(lmax, __shfl_xor(lmax, off, 32));
    }
    if (lane == 0) {
        atomicMin(&mm[b * 2 + 0], fkey(lmin));
        atomicMax(&mm[b * 2 + 1], fkey(lmax));
    }
}

__global__ void minmax_convert_kernel(const unsigned* __restrict__ mm, float* __restrict__ out) {
    int t = blockIdx.x * blockDim.x + threadIdx.x;
    if (t < B * 2) out[t] = funkey(mm[t]);
}

// pass 2: recompute tiles, bin into one block-shared LDS histogram, single flush
__global__ void sim_hist_kernel(const float* __restrict__ ui, const float* __restrict__ uj,
                                const float* __restrict__ mnmx, float* __restrict__ feat) {
    __shared__ unsigned hloc[BINS];
    int tid = threadIdx.x;
    int lane = tid & 31, wave = tid >> 5;
    if (tid < BINS) hloc[tid] = 0u;
    __syncthreads();
    int half = lane >> 4, l = lane & 15;
    int b = blockIdx.z;
    int row0 = blockIdx.x * 16;
    int col0 = (blockIdx.y * 8 + wave) * 16;
    const float* uib = ui + (long)b * N * F3;
    const float* ujb = uj + (long)b * N * F3;
    v8f c = sim_tile(uib, ujb, row0, col0, half, l);
    float mn = mnmx[b * 2 + 0], mx = mnmx[b * 2 + 1];
    float scale = (float)BINS / fmaxf(mx - mn, 1e-12f);
#pragma unroll
    for (int r = 0; r < 8; ++r) {
        float sg = sigmoidf_(c[r]);
        int idx = (int)((sg - mn) * scale);
        idx = idx < 0 ? 0 : (idx > BINS - 1 ? BINS - 1 : idx);
        atomicAdd(&hloc[idx], 1u);
    }
    __syncthreads();
    if (tid < BINS && hloc[tid] != 0u)
        atomicAdd(&feat[b * (K + BINS) + K + tid], (float)hloc[tid]);
}

// ---------------- tiny MLP head ----------------
__global__ void mlp_kernel(const float* __restrict__ feat,
                           const float* __restrict__ mw1, const float* __restrict__ mb1,
                           const float* __restrict__ mw2, const float* __restrict__ mb2,
                           const float* __restrict__ mw3, const float* __restrict__ mb3,
                           const float* __restrict__ mw4, const float* __restrict__ mb4,
                           const float* __restrict__ sw, const float* __restrict__ sb,
                           float* __restrict__ out) {
    int b = blockIdx.x, t = threadIdx.x;         // 32 threads
    __shared__ float f0[32], f1[M1], f2[M2], f3[M3], f4[M4];
    f0[t] = feat[b * 32 + t];
    __syncthreads();
    { float a = mb1[t]; for (int i = 0; i < 32; ++i) a += f0[i] * mw1[i * M1 + t]; f1[t] = fmaxf(a, 0.0f); }
    __syncthreads();
    if (t < M2) { float a = mb2[t]; for (int i = 0; i < M1; ++i) a += f1[i] * mw2[i * M2 + t]; f2[t] = fmaxf(a, 0.0f); }
    __syncthreads();
    if (t < M3) { float a = mb3[t]; for (int i = 0; i < M2; ++i) a += f2[i] * mw3[i * M3 + t]; f3[t] = fmaxf(a, 0.0f); }
    __syncthreads();
    if (t < M4) { float a = mb4[t]; for (int i = 0; i < M3; ++i) a += f3[i] * mw4[i * M4 + t]; f4[t] = fmaxf(a, 0.0f); }
    __syncthreads();
    if (t == 0) { float a = sb[0]; for (int i = 0; i < M4; ++i) a += f4[i] * sw[i]; out[b] = a; }
}

// ---------------- host orchestration ----------------
extern "C" void kernel_launch(void* const* d_in, const int* in_sizes, int n_in,
                              void* d_out, int out_size, void* d_ws, size_t ws_size,
                              hipStream_t stream) {
    const float* x_i  = (const float*)d_in[0];
    const int*   ei_i = (const int*)d_in[1];
    const float* x_j  = (const float*)d_in[2];
    const int*   ei_j = (const int*)d_in[3];
    const float* w1 = (const float*)d_in[4],  *b1 = (const float*)d_in[5];
    const float* w2 = (const float*)d_in[6],  *b2 = (const float*)d_in[7];
    const float* w3 = (const float*)d_in[8],  *b3 = (const float*)d_in[9];
    const float* attn_w = (const float*)d_in[10];
    const float* ntn_w  = (const float*)d_in[11];
    const float* ntn_v  = (const float*)d_in[12];
    const float* ntn_b  = (const float*)d_in[13];
    const float* mw1 = (const float*)d_in[14], *mb1 = (const float*)d_in[15];
    const float* mw2 = (const float*)d_in[16], *mb2 = (const float*)d_in[17];
    const float* mw3 = (const float*)d_in[18], *mb3 = (const float*)d_in[19];
    const float* mw4 = (const float*)d_in[20], *mb4 = (const float*)d_in[21];
    const float* sw = (const float*)d_in[22],  *sb = (const float*)d_in[23];
    float* out = (float*)d_out;

    // workspace layout (floats) — one dense A reused across the two graphs
    float* W = (float*)d_ws;
    float* A    = W;                                  // B*N*N
    float* t0   = A   + (long)B * N * N;              // B*N*F1
    float* t1   = t0  + (long)B * N * F1;             // B*N*F1
    float* u_i  = t1  + (long)B * N * F1;             // B*N*F3
    float* u_j  = u_i + (long)B * N * F3;             // B*N*F3
    float* dinv = u_j + (long)B * N * F3;             // B*N
    float* h_i  = dinv + (long)B * N;                 // B*16
    float* h_j  = h_i + B * 16;                       // B*16
    float* feat = h_j + B * 16;                       // B*32
    unsigned* mm = (unsigned*)(feat + B * 32);        // B*2
    float* mnmx = (float*)(mm + B * 2);               // B*2

    const long nA  = (long)B * N * N;
    const int  TB  = 256;

    for (int g = 0; g < 2; ++g) {
        const float* x  = g ? x_j : x_i;
        const int*   ei = g ? ei_j : ei_i;
        float*       u  = g ? u_j : u_i;
        float*       h  = g ? h_j : h_i;

        // dense normalized adjacency
        hipMemsetAsync(A, 0, nA * sizeof(float), stream);
        {
            long tot = (long)B * (E + N);
            adj_scatter_kernel<<<dim3((unsigned)((tot + TB - 1) / TB)), TB, 0, stream>>>(ei, A);
        }
        rowsum_dinv_kernel<<<dim3((B * N + TB - 1) / TB), TB, 0, stream>>>(A, dinv);
        adj_normalize_kernel<<<dim3((unsigned)((nA + TB - 1) / TB)), TB, 0, stream>>>(A, dinv);

        // layer 1: relu(A @ (x@w1) + b1)
        {
            long tot = (long)B * N * F1;
            xw_kernel<<<dim3((unsigned)((tot + TB - 1) / TB)), TB, 0, stream>>>(x, w1, t0, F_IN, F1);
            gcn_agg_wmma_kernel<4><<<dim3(N / 16 / 8, B), 256, 0, stream>>>(A, t0, b1, t1, 1);
        }
        // layer 2: relu(A @ (h@w2) + b2)
        {
            long tot = (long)B * N * F2;
            xw_kernel<<<dim3((unsigned)((tot + TB - 1) / TB)), TB, 0, stream>>>(t1, w2, t0, F1, F2);
            gcn_agg_wmma_kernel<2><<<dim3(N / 16 / 8, B), 256, 0, stream>>>(A, t0, b2, t1, 1);
        }
        // layer 3: A @ (h@w3) + b3 (no relu)
        {
            long tot = (long)B * N * F3;
            xw_kernel<<<dim3((unsigned)((tot + TB - 1) / TB)), TB, 0, stream>>>(t1, w3, t0, F2, F3);
            gcn_agg_wmma_kernel<1><<<dim3(N / 16 / 8, B), 256, 0, stream>>>(A, t0, b3, u, 0);
        }
        att_pool_kernel<<<dim3(B), 256, 0, stream>>>(u, attn_w, h);
    }

    // NTN interaction (also zeroes the histogram slots of feat)
    ntn_kernel<<<dim3(B), 32, 0, stream>>>(h_i, h_j, ntn_w, ntn_v, ntn_b, feat);

    // similarity histogram: pass 1 (min/max), convert, pass 2 (bin counts)
    minmax_init_kernel<<<dim3((B + 63) / 64), 64, 0, stream>>>(mm);
    sim_minmax_kernel<<<dim3(N / 16, N / 16 / 8, B), 256, 0, stream>>>(u_i, u_j, mm);
    minmax_convert_kernel<<<dim3((B * 2 + 63) / 64), 64, 0, stream>>>(mm, mnmx);
    sim_hist_kernel<<<dim3(N / 16, N / 16 / 8, B), 256, 0, stream>>>(u_i, u_j, mnmx, feat);

    // scoring MLP
    mlp_kernel<<<dim3(B), 32, 0, stream>>>(feat, mw1, mb1, mw2, mb2, mw3, mb3,
                                           mw4, mb4, sw, sb, out);
    (void)in_sizes; (void)n_in; (void)out_size; (void)ws_size;
}